// Reblurring_Module_16269336117255
// MI455X (gfx1250) — compile-verified
//
#include <hip/hip_runtime.h>
#include <cstdint>
#include <cstddef>

// ---- problem constants: N,C,H,W = 16,3,256,256 ; K=5 -> 25 per-pixel taps ----
#define HGT 256
#define WID 256
#define S25 25
#define TW  32          // output tile width
#define TH  16          // output tile height
#define PKX 48          // perpix LDS tile x extent (left halo 8 for alignment, covers [tx0-8, tx0+40))
#define PKY 26          // perpix LDS tile y extent (halo 5 top/bottom: 16+10)
#define STX 42          // stats region x: mask region + conv halo 3 (36+6)
#define STY 26          // 20+6
#define MKX 36          // mask region x: out + fac halo 2 (32+4)
#define MKY 20          // 16+4

// ---- LDS partition (in floats) ----
#define OFF_PK    0
#define OFF_ST    (S25*PKY*PKX)                 // 31200 : perpix tile [25][26][48]
#define OFF_W     (OFF_ST + 2*STY*STX)          // +2184 : stats [2][26][42]
#define OFF_MASK  (OFF_W + 104)                 // +104  : conv weights (98, padded)
#define OFF_MSK   (OFF_MASK + MKY*MKX)          // +720  : mask [20][36]
#define OFF_BLUR  (OFF_MSK + 3*MKY*MKX)         // +2160 : masked feat [3][20][36]
#define SMEM_FLTS (OFF_BLUR + 3*MKY*MKX)        // +2160 : blur tile [3][20][36]
#define SMEM_BYTES (SMEM_FLTS * 4)              // = 154112 B  (2 WGs / 320KB WGP)

typedef __attribute__((ext_vector_type(4))) unsigned v4u_t;
typedef __attribute__((ext_vector_type(8))) int      v8i_t;
typedef __attribute__((ext_vector_type(4))) int      v4i_t;

__global__ __launch_bounds__(256)
void reblur_fused_kernel(const float* __restrict__ blur,
                         const float* __restrict__ pk0,
                         const float* __restrict__ pk1,
                         const float* __restrict__ pk2,
                         const float* __restrict__ w0,
                         const float* __restrict__ w1,
                         const float* __restrict__ w2,
                         float* __restrict__ out)
{
    extern __shared__ float smem[];
    float* sh_pk   = smem + OFF_PK;
    float* sh_st   = smem + OFF_ST;
    float* sh_w    = smem + OFF_W;
    float* sh_mask = smem + OFF_MASK;
    float* sh_msk  = smem + OFF_MSK;
    float* sh_blur = smem + OFF_BLUR;

    const int tid = (int)threadIdx.x;
    const int tx0 = (int)blockIdx.x * TW;
    const int ty0 = (int)blockIdx.y * TH;
    const int n   = (int)blockIdx.z;
    const int ox  = (tx0 - 8 > 0) ? tx0 - 8 : 0;   // clamped perpix-tile origin
    const int oy  = (ty0 - 5 > 0) ? ty0 - 5 : 0;

    // wave-uniform "am I wave 0" flag (readfirstlane -> scalar branch; tensor ops ignore EXEC)
    const bool wave0 = ((__builtin_amdgcn_readfirstlane((int)threadIdx.x) >> 5) == 0);

    // ------------------------------------------------------------------
    // Blur tile (3ch, mask region + fac halo): zero-init, then async-to-LDS
    // for in-image lanes. ds_store -> s_wait_dscnt -> async (ASYNCcnt path).
    // ------------------------------------------------------------------
    for (int idx = tid; idx < 3*MKY*MKX; idx += 256) sh_blur[idx] = 0.0f;
    asm volatile("s_wait_dscnt 0" ::: "memory");
    for (int idx = tid; idx < 3*MKY*MKX; idx += 256) {
        int c  = idx / (MKY*MKX);
        int r  = idx - c * (MKY*MKX);
        int my = r / MKX, mx = r - my * MKX;
        int gy = ty0 - 2 + my, gx = tx0 - 2 + mx;
        if (gy >= 0 && gy < HGT && gx >= 0 && gx < WID) {
            unsigned lo = (unsigned)(uintptr_t)(void*)&sh_blur[idx];
            unsigned long long ga = (unsigned long long)(uintptr_t)
                (blur + (((size_t)(n*3 + c) << 16) + ((size_t)gy << 8) + (size_t)gx));
            asm volatile("global_load_async_to_lds_b32 %0, %1, off"
                         :: "v"(lo), "v"(ga) : "memory");
        }
    }
    asm volatile("s_wait_asynccnt 0" ::: "memory");

    float acc[2][3] = {};

    for (int b = 0; b < 3; ++b) {
        const float* pkb = (b == 0) ? pk0 : ((b == 1) ? pk1 : pk2);
        const float* wb  = (b == 0) ? w0  : ((b == 1) ? w1  : w2);

        __syncthreads();   // previous branch done reading sh_pk / sh_w

        // ---------------- TDM: perpix tile [s=25][y=26][x=48] -> LDS ----------------
        if (wave0) {
            unsigned long long ga = (unsigned long long)(uintptr_t)pkb
                + 4ull * (((size_t)(n * S25) << 16) + ((size_t)oy << 8) + (size_t)ox);
            unsigned lds_off = (unsigned)(uintptr_t)(void*)sh_pk;
            unsigned td0 = (unsigned)(WID - ox);    // remaining x extent (right clip -> 0-fill)
            unsigned td1 = (unsigned)(HGT - oy);    // remaining y extent (bottom clip)
            // D# group0: count=1 | lds_addr | global_addr[56:0] | type=2
            v4u_t g0;
            g0[0] = 1u;
            g0[1] = lds_off;
            g0[2] = (unsigned)ga;
            g0[3] = ((unsigned)(ga >> 32) & 0x01FFFFFFu) | (2u << 30);
            // D# group1: data_size=4B | dims/tiles | strides (y=256, s=65536)
            v8i_t g1;
            g1[0] = (int)(2u << 16);                                   // data_size=2 (4B), wg_mask=0
            g1[1] = (int)((td0 & 0xFFFFu) << 16);                      // tensor_dim0[15:0]
            g1[2] = (int)((td0 >> 16) | ((td1 & 0xFFFFu) << 16));      // td0[31:16] | td1[15:0]
            g1[3] = (int)((td1 >> 16) | ((unsigned)PKX << 16));        // td1[31:16] | tile_dim0=48
            g1[4] = (int)((unsigned)PKY | ((unsigned)S25 << 16));      // tile_dim1=26 | tile_dim2=25
            g1[5] = 256;                                               // tensor_dim0_stride lo
            g1[6] = 0;                                                 // stride0 hi | stride1[15:0]
            g1[7] = 1;                                                 // stride1[47:16] = 65536>>16
            v4i_t g2; g2[0] = S25; g2[1] = 0; g2[2] = 0; g2[3] = 0;    // tensor_dim2=25
            v4i_t g3; g3[0] = 0;   g3[1] = 0; g3[2] = 0; g3[3] = 0;
            v8i_t g4; g4[0]=0; g4[1]=0; g4[2]=0; g4[3]=0;              // unused extended group
            g4[4]=0; g4[5]=0; g4[6]=0; g4[7]=0;
            __builtin_amdgcn_tensor_load_to_lds(g0, g1, g2, g3, g4, 0);
        }
        if (tid < 98) sh_w[tid] = wb[tid];          // 7x7x2 conv weights -> LDS
        if (wave0) __builtin_amdgcn_s_wait_tensorcnt((short)0);
        __syncthreads();                            // perpix tile + weights visible

        // ---------------- stats: mean & max over 25 taps, zero outside image ----------------
        for (int idx = tid; idx < STY*STX; idx += 256) {
            int sy = idx / STX, sx = idx - sy * STX;
            int gy = ty0 - 5 + sy, gx = tx0 - 5 + sx;
            float av = 0.0f, mxv = 0.0f;
            if (gy >= 0 && gy < HGT && gx >= 0 && gx < WID) {
                int ry = gy - oy, rx = gx - ox;
                float sum = 0.0f, m = -3.4e38f;
                #pragma unroll
                for (int s = 0; s < S25; ++s) {
                    float v = sh_pk[(s * PKY + ry) * PKX + rx];
                    sum += v;
                    m = fmaxf(m, v);
                }
                av = sum * (1.0f / 25.0f);
                mxv = m;
            }
            sh_st[idx]           = av;   // channel 0: avg
            sh_st[STY*STX + idx] = mxv;  // channel 1: max
        }
        __syncthreads();

        // ---------------- 7x7 conv + sigmoid -> mask ; masked = blur * mask ----------------
        for (int idx = tid; idx < MKY*MKX; idx += 256) {
            int my = idx / MKX, mx = idx - my * MKX;
            float a = 0.0f;
            #pragma unroll
            for (int ci = 0; ci < 2; ++ci) {
                #pragma unroll
                for (int dy = 0; dy < 7; ++dy) {
                    const float* srow = &sh_st[(ci * STY + my + dy) * STX + mx];
                    const float* wrow = &sh_w[(ci * 7 + dy) * 7];
                    #pragma unroll
                    for (int dx = 0; dx < 7; ++dx)
                        a = fmaf(srow[dx], wrow[dx], a);
                }
            }
            float m = 1.0f / (1.0f + __expf(-a));
            sh_mask[idx] = m;
            #pragma unroll
            for (int c = 0; c < 3; ++c)
                sh_msk[c * MKY*MKX + idx] = sh_blur[c * MKY*MKX + idx] * m;
        }
        __syncthreads();

        // ---------------- fac (per-pixel 5x5 adaptive filter) + sharp, accumulate ----------------
        #pragma unroll
        for (int it = 0; it < 2; ++it) {
            int idx = tid + it * 256;
            int y = idx >> 5, x = idx & 31;
            int ry = ty0 + y - oy, rx = tx0 + x - ox;
            float pkv[S25];
            #pragma unroll
            for (int s = 0; s < S25; ++s)
                pkv[s] = sh_pk[(s * PKY + ry) * PKX + rx];
            float m = sh_mask[(y + 2) * MKX + (x + 2)];
            #pragma unroll
            for (int c = 0; c < 3; ++c) {
                float a = sh_blur[(c * MKY + y + 2) * MKX + (x + 2)] * (1.0f - m);
                #pragma unroll
                for (int i = 0; i < 5; ++i)
                    #pragma unroll
                    for (int j = 0; j < 5; ++j)
                        a = fmaf(sh_msk[(c * MKY + y + i) * MKX + (x + j)],
                                 pkv[i * 5 + j], a);
                acc[it][c] += a;
            }
        }
    }

    // ---------------- single coalesced write of the 3-branch mean ----------------
    #pragma unroll
    for (int it = 0; it < 2; ++it) {
        int idx = tid + it * 256;
        int y = idx >> 5, x = idx & 31;
        int gy = ty0 + y, gx = tx0 + x;
        #pragma unroll
        for (int c = 0; c < 3; ++c)
            out[((size_t)(n * 3 + c) << 16) + ((size_t)gy << 8) + (size_t)gx]
                = acc[it][c] * (1.0f / 3.0f);
    }
}

extern "C" void kernel_launch(void* const* d_in, const int* in_sizes, int n_in,
                              void* d_out, int out_size, void* d_ws, size_t ws_size,
                              hipStream_t stream)
{
    const float* blur = (const float*)d_in[0];
    const float* pk0  = (const float*)d_in[1];
    const float* pk1  = (const float*)d_in[2];
    const float* pk2  = (const float*)d_in[3];
    const float* w0   = (const float*)d_in[4];
    const float* w1   = (const float*)d_in[5];
    const float* w2   = (const float*)d_in[6];
    float* outp = (float*)d_out;

    const int N = in_sizes[0] / (3 * HGT * WID);   // = 16

    // dynamic LDS 154 KB > 64 KB default: raise the cap (idempotent, capture-safe)
    (void)hipFuncSetAttribute((const void*)reblur_fused_kernel,
                              hipFuncAttributeMaxDynamicSharedMemorySize, SMEM_BYTES);

    dim3 grid(WID / TW, HGT / TH, N);   // (8, 16, 16) tiles
    dim3 block(256);                    // 8 wave32s
    reblur_fused_kernel<<<grid, block, SMEM_BYTES, stream>>>(
        blur, pk0, pk1, pk2, w0, w1, w2, outp);
}